// Transformer_7164005449941
// MI455X (gfx1250) — compile-verified
//
#include <hip/hip_runtime.h>
#include <math.h>

typedef float v2f __attribute__((ext_vector_type(2)));
typedef float v8f __attribute__((ext_vector_type(8)));

// xor-shuffle within wave32 via ds_swizzle (and=0x1f, or=0, xor=m)
#define SWZ_XOR1(v) __int_as_float(__builtin_amdgcn_ds_swizzle(__float_as_int(v), 0x041f))
#define SWZ_XOR2(v) __int_as_float(__builtin_amdgcn_ds_swizzle(__float_as_int(v), 0x081f))
#define SWZ_XOR4(v) __int_as_float(__builtin_amdgcn_ds_swizzle(__float_as_int(v), 0x101f))
#define SWZ_XOR8(v) __int_as_float(__builtin_amdgcn_ds_swizzle(__float_as_int(v), 0x201f))

// single-instruction v_rcp_f32 (~1 ulp) instead of IEEE div sequence
__device__ __forceinline__ float fast_rcp(float x) { return __builtin_amdgcn_rcpf(x); }

// ---------------------------------------------------------------- zero
__global__ __launch_bounds__(256) void k_zero(float* __restrict__ p, int n) {
    int i = blockIdx.x * 256 + threadIdx.x;
    if (i < n) p[i] = 0.f;
}

// ---------------------------------------------------------------- P = nf @ (Wq@Wdot) / 64
__global__ __launch_bounds__(256) void k_p(const float* __restrict__ nf,
                                           const float* __restrict__ Wq,
                                           const float* __restrict__ Wdot,
                                           float* __restrict__ P, int N) {
    int n = blockIdx.x * 256 + threadIdx.x;
    if (n >= N) return;
    float x[16];
    const float4* xr = (const float4*)(nf + (size_t)n * 16);
#pragma unroll
    for (int i = 0; i < 4; ++i) {
        float4 v = xr[i];
        x[4*i] = v.x; x[4*i+1] = v.y; x[4*i+2] = v.z; x[4*i+3] = v.w;
    }
    float q[16];
#pragma unroll
    for (int i = 0; i < 16; ++i) {
        float a = 0.f;
#pragma unroll
        for (int c = 0; c < 16; ++c) a = fmaf(x[c], Wq[c*16 + i], a);
        q[i] = a;
    }
    float* pr = P + (size_t)n * 16;
#pragma unroll
    for (int j = 0; j < 16; ++j) {
        float a = 0.f;
#pragma unroll
        for (int i = 0; i < 16; ++i) a = fmaf(q[i], Wdot[i*16 + j], a);
        pr[j] = a * (1.f / 64.f);   // (1/sqrt16 from Wq) * (1/M0)
    }
}

// ---------------------------------------------------------------- Pass A: WMMA edge GEMMs
// 1 wave = 16 edges. t[e,o] = Z[e,:] @ W2r[:,o], Z[e,h*16+c] = hk/hv[e,h]*f[e,c].
// k path needs only cols 0..15 (logits); v path needs cols 0..27.
__global__ __launch_bounds__(256) void passA_kernel(
    const float* __restrict__ nf,   const float* __restrict__ sh,
    const float* __restrict__ cut,  const float* __restrict__ scal,
    const float* __restrict__ W1k,  const float* __restrict__ W2k,
    const float* __restrict__ W1v,  const float* __restrict__ W2v,
    const int* __restrict__ esrc,   const int* __restrict__ edst,
    const float* __restrict__ P,    float* __restrict__ tv,
    float* __restrict__ exps,       float* __restrict__ zbuf, int E)
{
    // column-major, stride 258 (even -> 8B-aligned b64 reads, bank-spread)
    __shared__ float sW2k[16 * 258];
    __shared__ float sW2v[32 * 258];   // cols 28..31 zero-padded
    const int t = threadIdx.x;
    for (int i = t; i < 16 * 256; i += 256) {
        int oo = i >> 8, r = i & 255, h = r >> 4, c = r & 15;
        sW2k[oo * 258 + r] = W2k[h * 448 + c * 16 + oo] * 0.0625f;
    }
    for (int i = t; i < 32 * 256; i += 256) {
        int oo = i >> 8, r = i & 255, h = r >> 4, c = r & 15;
        float w = 0.f;
        if (oo < 16)      w = W2v[h * 448 + c * 16 + oo];
        else if (oo < 24) w = W2v[h * 448 + 256 + c * 8 + (oo - 16)];
        else if (oo < 28) w = W2v[h * 448 + 384 + c * 4 + (oo - 24)];
        sW2v[oo * 258 + r] = w * 0.0625f;
    }
    __syncthreads();

    const int lane = t & 31;
    const int wave = t >> 5;
    const int half = lane >> 4;      // K-half of the wave
    const int o    = lane & 15;      // N column / edge-in-tile for A rows
    const int tile = blockIdx.x * 8 + wave;
    if (tile * 16 >= E) return;      // wave-uniform
    const int ebase = tile * 16;
    const int eme   = ebase + o;

    // per-lane f slice: this K-half only needs c = 4q + 2*half + {0,1}
    const int   srci = esrc[eme];
    const float* frow = nf + (size_t)srci * 16 + 2 * half;
    float fa[8];
#pragma unroll
    for (int q = 0; q < 4; ++q) {
        float2 f2 = *(const float2*)(frow + 4 * q);
        fa[2*q] = f2.x; fa[2*q+1] = f2.y;
    }
    float s10[10];
    const float* srow = scal + (size_t)eme * 10;
#pragma unroll
    for (int b = 0; b < 10; ++b) s10[b] = srow[b];

    const float r10 = 0.31622776601683794f;  // 1/sqrt(10)
    float hk[16], hv[16];
#pragma unroll
    for (int j = 0; j < 16; ++j) {
        float uk = 0.f, uv = 0.f;
#pragma unroll
        for (int b = 0; b < 10; ++b) {        // uniform addrs -> s_loads
            uk = fmaf(s10[b], W1k[b * 16 + j], uk);
            uv = fmaf(s10[b], W1v[b * 16 + j], uv);
        }
        uk *= r10; uv *= r10;
        hk[j] = uk * fast_rcp(1.f + __expf(-uk));   // silu, v_rcp_f32
        hv[j] = uv * fast_rcp(1.f + __expf(-uv));
    }

    v8f ak0 = {0.f,0.f,0.f,0.f,0.f,0.f,0.f,0.f};
    v8f av0 = ak0, av1 = ak0;
    const float* bk  = sW2k + o * 258 + 2 * half;
    const float* bv0 = sW2v + o * 258 + 2 * half;
    const float* bv1 = sW2v + (16 + o) * 258 + 2 * half;

#pragma unroll
    for (int step = 0; step < 64; ++step) {   // K = 256, 4 per WMMA
        const int hi = step >> 2;             // compile-time indices only
        const int q  = step & 3;
        v2f Ak, Av;
        Ak[0] = fa[2*q]   * hk[hi];
        Ak[1] = fa[2*q+1] * hk[hi];
        Av[0] = fa[2*q]   * hv[hi];
        Av[1] = fa[2*q+1] * hv[hi];
        v2f Bk  = *(const v2f*)(bk  + 4 * step);
        v2f Bv0 = *(const v2f*)(bv0 + 4 * step);
        v2f Bv1 = *(const v2f*)(bv1 + 4 * step);
        ak0 = __builtin_amdgcn_wmma_f32_16x16x4_f32(false, Ak, false, Bk,  (short)0, ak0, false, false);
        av0 = __builtin_amdgcn_wmma_f32_16x16x4_f32(false, Av, false, Bv0, (short)0, av0, false, false);
        av1 = __builtin_amdgcn_wmma_f32_16x16x4_f32(false, Av, false, Bv1, (short)0, av1, false, false);
    }

    // ---- logits from ak0: D[e = v+8*half][o = lane&15]
#pragma unroll
    for (int v = 0; v < 8; ++v) {
        const int ee = ebase + 8 * half + v;       // uniform within half
        const int dn = edst[ee];
        float pv = P[(size_t)dn * 16 + o];
        float r  = ak0[v] * pv;
        r += SWZ_XOR1(r); r += SWZ_XOR2(r); r += SWZ_XOR4(r); r += SWZ_XOR8(r);
        float sh0 = sh[(size_t)ee * 9];
        float ex  = cut[ee] * __expf(sh0 * r);
        if (o == 0) {
            exps[ee] = ex;
            unsafeAtomicAdd(zbuf + dn, ex);
        }
    }

    // ---- spill t_v in [tile][col][e] layout (float4 from accumulator VGPRs)
    float* tvb = tv + (size_t)tile * 448 + 8 * half;
    *(float4*)(tvb + o * 16)     = make_float4(av0[0], av0[1], av0[2], av0[3]);
    *(float4*)(tvb + o * 16 + 4) = make_float4(av0[4], av0[5], av0[6], av0[7]);
    if (o < 12) {
        *(float4*)(tvb + (16 + o) * 16)     = make_float4(av1[0], av1[1], av1[2], av1[3]);
        *(float4*)(tvb + (16 + o) * 16 + 4) = make_float4(av1[4], av1[5], av1[6], av1[7]);
    }
}

// ---------------------------------------------------------------- Pass B: alpha + scatter
__global__ __launch_bounds__(256) void passB_kernel(
    const float* __restrict__ tv,  const float* __restrict__ exps,
    const float* __restrict__ zbuf, const float* __restrict__ sh,
    const int* __restrict__ edst,  float* __restrict__ facc, int E)
{
    int e = blockIdx.x * 256 + threadIdx.x;
    if (e >= E) return;
    const int tile = e >> 4, el = e & 15;
    const float* tb = tv + (size_t)tile * 448 + el;
    float tt[28];
#pragma unroll
    for (int oo = 0; oo < 28; ++oo) tt[oo] = tb[oo * 16];  // coalesced per oo
    const int dn = edst[e];
    float zz = zbuf[dn];
    zz = (zz == 0.f) ? 1.f : zz;
    float alpha = exps[e] * fast_rcp(zz);
    float w = sqrtf(fmaxf(alpha, 0.f));
    float s9[9];
#pragma unroll
    for (int m = 0; m < 9; ++m) s9[m] = sh[(size_t)e * 9 + m];
    float* fr = facc + (size_t)dn * 60;
#pragma unroll
    for (int oo = 0; oo < 16; ++oo)
        unsafeAtomicAdd(fr + oo, w * tt[oo] * s9[0]);
#pragma unroll
    for (int oo = 0; oo < 8; ++oo)
#pragma unroll
        for (int m = 0; m < 3; ++m)
            unsafeAtomicAdd(fr + 16 + oo * 3 + m, w * tt[16 + oo] * s9[1 + m]);
#pragma unroll
    for (int oo = 0; oo < 4; ++oo)
#pragma unroll
        for (int m = 0; m < 5; ++m)
            unsafeAtomicAdd(fr + 40 + oo * 5 + m, w * tt[24 + oo] * s9[4 + m]);
}

// ---------------------------------------------------------------- Pass C: row-normalize
__global__ __launch_bounds__(256) void passC_kernel(const float* __restrict__ facc,
                                                    float* __restrict__ out, int N)
{
    int n = blockIdx.x * 256 + threadIdx.x;
    if (n >= N) return;
    const float4* r4 = (const float4*)(facc + (size_t)n * 60);
    float4* o4 = (float4*)(out + (size_t)n * 60);
    float buf[60];
    float ss = 0.f;
#pragma unroll
    for (int i = 0; i < 15; ++i) {
        float4 v = r4[i];
        buf[4*i] = v.x; buf[4*i+1] = v.y; buf[4*i+2] = v.z; buf[4*i+3] = v.w;
        ss += v.x*v.x + v.y*v.y + v.z*v.z + v.w*v.w;
    }
    float inv = 1.f / fmaxf(sqrtf(ss), 1e-12f);
#pragma unroll
    for (int i = 0; i < 15; ++i)
        o4[i] = make_float4(buf[4*i]*inv, buf[4*i+1]*inv, buf[4*i+2]*inv, buf[4*i+3]*inv);
}

// ---------------------------------------------------------------- host
extern "C" void kernel_launch(void* const* d_in, const int* in_sizes, int n_in,
                              void* d_out, int out_size, void* d_ws, size_t ws_size,
                              hipStream_t stream)
{
    const float* nf   = (const float*)d_in[0];
    const float* sh   = (const float*)d_in[1];
    const float* cut  = (const float*)d_in[2];
    const float* scal = (const float*)d_in[3];
    const float* Wq   = (const float*)d_in[4];
    const float* W1k  = (const float*)d_in[5];
    const float* W2k  = (const float*)d_in[6];
    const float* W1v  = (const float*)d_in[7];
    const float* W2v  = (const float*)d_in[8];
    const float* Wdot = (const float*)d_in[9];
    const int*   esrc = (const int*)d_in[10];
    const int*   edst = (const int*)d_in[11];

    const int N = in_sizes[0] / 16;
    const int E = in_sizes[2];

    float* base = (float*)d_ws;
    size_t off = 0;
    auto alloc = [&](size_t nfl) {
        float* p = base + off;
        off += (nfl + 3) & ~(size_t)3;   // keep 16B alignment
        return p;
    };
    float* zbuf = alloc((size_t)N);            // z[N]
    float* facc = alloc((size_t)N * 60);       // accumulated output (contiguous with z)
    float* exps = alloc((size_t)E);            // per-edge exp
    float* Pm   = alloc((size_t)N * 16);       // P = nf @ (Wq@Wdot)/64
    float* tv   = alloc((size_t)E * 28);       // per-edge t_v

    const int zcount = (int)((facc - zbuf) + (size_t)N * 60);
    k_zero<<<(zcount + 255) / 256, 256, 0, stream>>>(zbuf, zcount);
    k_p<<<(N + 255) / 256, 256, 0, stream>>>(nf, Wq, Wdot, Pm, N);

    const int tiles = E / 16;                  // E is a multiple of 16
    passA_kernel<<<(tiles + 7) / 8, 256, 0, stream>>>(nf, sh, cut, scal, W1k, W2k,
                                                      W1v, W2v, esrc, edst, Pm,
                                                      tv, exps, zbuf, E);
    passB_kernel<<<(E + 255) / 256, 256, 0, stream>>>(tv, exps, zbuf, sh, edst, facc, E);
    passC_kernel<<<(N + 255) / 256, 256, 0, stream>>>(facc, (float*)d_out, N);
}